// GraphCON_GCN_65292092834213
// MI455X (gfx1250) — compile-verified
//
#include <hip/hip_runtime.h>

// ---------------- problem constants ----------------
#define NNODES   50000
#define NEDGES   800000
#define NFEAT    16
#define NHID     64
#define NGRAPHS  500
#define NLAYERS  5
#define NPAD     50176          // 392 * 128, padded row count so WMMA tiles need no guards
#define DT_C     1.0f
#define ALPHA_C  1.0f
#define GAMMA_C  1.0f

typedef float v2f __attribute__((ext_vector_type(2)));
typedef float v8f __attribute__((ext_vector_type(8)));

__device__ __forceinline__ void atomic_add_f32(float* p, float v) {
    // non-returning relaxed agent-scope fp32 atomic -> global_atomic_add_f32 (no CAS loop)
    __hip_atomic_fetch_add(p, v, __ATOMIC_RELAXED, __HIP_MEMORY_SCOPE_AGENT);
}

// ---------------- init: zero degree buffer + output ----------------
__global__ void gcn_zero_kernel(float* __restrict__ deg, float* __restrict__ out) {
    int i = blockIdx.x * blockDim.x + threadIdx.x;
    if (i < NPAD) deg[i] = 0.0f;
    if (i < NGRAPHS) out[i] = 0.0f;
}

// ---------------- degree (layer-invariant) ----------------
__global__ void gcn_degree_kernel(const int* __restrict__ dst, float* __restrict__ deg) {
    int e = blockIdx.x * blockDim.x + threadIdx.x;
    if (e < NEDGES) atomic_add_f32(&deg[dst[e]], 1.0f);
}

__global__ void gcn_dinv_kernel(float* __restrict__ deg) {
    int i = blockIdx.x * blockDim.x + threadIdx.x;
    if (i < NPAD) deg[i] = __frsqrt_rn(deg[i] + 1.0f);   // +1 = self loop
}

// ---------------- encoder: Y = [x|pos] @ enc_W + enc_b ; X = Y  (WMMA f32) ----------------
__global__ void __launch_bounds__(256)
gcn_encoder_kernel(const float* __restrict__ x, const float* __restrict__ pos,
                   const float* __restrict__ Wenc, const float* __restrict__ benc,
                   float* __restrict__ X, float* __restrict__ Y) {
    const int lane  = threadIdx.x & 31;
    const int wave  = threadIdx.x >> 5;
    const int row0  = blockIdx.x * 128 + wave * 16;
    const int m     = lane & 15;             // A row / B col within tile
    const int khalf = (lane >> 4) * 2;       // K offset for this half-wave

    v8f acc[4] = {};                         // 16 rows x 64 cols per wave

    const int rowc = min(row0 + m, NNODES - 1);   // clamp input gather only
    for (int k0 = 0; k0 < NFEAT; k0 += 4) {
        v2f a;
        {
            int c0 = k0 + khalf;
            int c1 = c0 + 1;
            a.x = (c0 < 14) ? x[rowc * 14 + c0] : pos[rowc * 2 + (c0 - 14)];
            a.y = (c1 < 14) ? x[rowc * 14 + c1] : pos[rowc * 2 + (c1 - 14)];
        }
#pragma unroll
        for (int t = 0; t < 4; ++t) {
            v2f b;
            b.x = Wenc[(k0 + khalf + 0) * NHID + t * 16 + m];
            b.y = Wenc[(k0 + khalf + 1) * NHID + t * 16 + m];
            acc[t] = __builtin_amdgcn_wmma_f32_16x16x4_f32(
                false, a, false, b, (short)0, acc[t], false, false);
        }
    }
#pragma unroll
    for (int t = 0; t < 4; ++t) {
        int col = t * 16 + m;
        float bias = benc[col];
#pragma unroll
        for (int v = 0; v < 8; ++v) {
            int row = row0 + v + ((lane >> 4) << 3);
            float val = acc[t][v] + bias;
            Y[row * NHID + col] = val;
            X[row * NHID + col] = val;
        }
    }
}

// ---- fused dual GEMM: hW = X@Wc ; agg = X@Wr + br + bc + hW*dinv(row)^2 (self loop) ----
__global__ void __launch_bounds__(256)
gcn_gemm_kernel(const float* __restrict__ X,
                const float* __restrict__ Wc, const float* __restrict__ Wr,
                const float* __restrict__ bc, const float* __restrict__ br,
                const float* __restrict__ dinv,
                float* __restrict__ hW, float* __restrict__ agg) {
    const int lane  = threadIdx.x & 31;
    const int wave  = threadIdx.x >> 5;
    const int row0  = blockIdx.x * 128 + wave * 16;
    const int m     = lane & 15;
    const int khalf = (lane >> 4) * 2;

    v8f accC[4] = {};
    v8f accR[4] = {};

    const float* arow = X + (size_t)(row0 + m) * NHID;
    for (int k0 = 0; k0 < NHID; k0 += 4) {
        // two contiguous, 8B-aligned K values per lane -> single global_load_b64
        const v2f a = *(const v2f*)(arow + k0 + khalf);
#pragma unroll
        for (int t = 0; t < 4; ++t) {
            v2f b0, b1;
            b0.x = Wc[(k0 + khalf + 0) * NHID + t * 16 + m];
            b0.y = Wc[(k0 + khalf + 1) * NHID + t * 16 + m];
            b1.x = Wr[(k0 + khalf + 0) * NHID + t * 16 + m];
            b1.y = Wr[(k0 + khalf + 1) * NHID + t * 16 + m];
            accC[t] = __builtin_amdgcn_wmma_f32_16x16x4_f32(
                false, a, false, b0, (short)0, accC[t], false, false);
            accR[t] = __builtin_amdgcn_wmma_f32_16x16x4_f32(
                false, a, false, b1, (short)0, accR[t], false, false);
        }
    }
#pragma unroll
    for (int t = 0; t < 4; ++t) {
        int col = t * 16 + m;
        float bias = br[col] + bc[col];
#pragma unroll
        for (int v = 0; v < 8; ++v) {
            int row = row0 + v + ((lane >> 4) << 3);
            float h  = accC[t][v];
            float dv = dinv[row];
            hW[row * NHID + col]  = h;
            agg[row * NHID + col] = accR[t][v] + bias + h * dv * dv;
        }
    }
}

// ---------------- edge scatter: agg[dst] += hW[src] * dinv[src]*dinv[dst] ----------------
__global__ void __launch_bounds__(256)
gcn_scatter_kernel(const int* __restrict__ src, const int* __restrict__ dst,
                   const float* __restrict__ hW, const float* __restrict__ dinv,
                   float* __restrict__ agg) {
    int e = blockIdx.x * (blockDim.x >> 5) + (threadIdx.x >> 5);
    if (e >= NEDGES) return;
    int lane = threadIdx.x & 31;
    int s = src[e];
    int d = dst[e];
    float norm = dinv[s] * dinv[d];
    float2 hv = ((const float2*)hW)[s * 32 + lane];     // lane -> 2 coalesced floats
    atomic_add_f32(&agg[d * NHID + 2 * lane + 0], hv.x * norm);
    atomic_add_f32(&agg[d * NHID + 2 * lane + 1], hv.y * norm);
}

// ---------------- ODE update: Y += DT*(relu(agg) - a*Y - g*X); X += DT*Y ----------------
__global__ void gcn_update_kernel(float* __restrict__ X, float* __restrict__ Y,
                                  const float* __restrict__ agg) {
    int i = blockIdx.x * blockDim.x + threadIdx.x;           // float4 index
    if (i >= (NNODES * NHID) / 4) return;
    float4 a  = ((const float4*)agg)[i];
    float4 y  = ((float4*)Y)[i];
    float4 xv = ((float4*)X)[i];
    float4 yn, xn;
    yn.x = y.x + DT_C * (fmaxf(a.x, 0.0f) - ALPHA_C * y.x - GAMMA_C * xv.x);
    yn.y = y.y + DT_C * (fmaxf(a.y, 0.0f) - ALPHA_C * y.y - GAMMA_C * xv.y);
    yn.z = y.z + DT_C * (fmaxf(a.z, 0.0f) - ALPHA_C * y.z - GAMMA_C * xv.z);
    yn.w = y.w + DT_C * (fmaxf(a.w, 0.0f) - ALPHA_C * y.w - GAMMA_C * xv.w);
    xn.x = xv.x + DT_C * yn.x;
    xn.y = xv.y + DT_C * yn.y;
    xn.z = xv.z + DT_C * yn.z;
    xn.w = xv.w + DT_C * yn.w;
    ((float4*)Y)[i] = yn;
    ((float4*)X)[i] = xn;
}

// ---------------- decoder + graph pooling ----------------
__global__ void __launch_bounds__(256)
gcn_decoder_kernel(const float* __restrict__ X, const float* __restrict__ decW,
                   const float* __restrict__ decb, const int* __restrict__ batch,
                   float* __restrict__ out) {
    int node = blockIdx.x * (blockDim.x >> 5) + (threadIdx.x >> 5);
    if (node >= NNODES) return;
    int lane = threadIdx.x & 31;
    float2 xv = ((const float2*)X)[node * 32 + lane];
    float2 wv = ((const float2*)decW)[lane];
    float v = xv.x * wv.x + xv.y * wv.y;
#pragma unroll
    for (int off = 16; off > 0; off >>= 1) v += __shfl_xor(v, off, 32);
    if (lane == 0) atomic_add_f32(&out[batch[node]], v + decb[0]);
}

// ---------------- host launcher ----------------
extern "C" void kernel_launch(void* const* d_in, const int* in_sizes, int n_in,
                              void* d_out, int out_size, void* d_ws, size_t ws_size,
                              hipStream_t stream) {
    const float* x     = (const float*)d_in[0];
    const float* pos   = (const float*)d_in[1];
    const int*   edge  = (const int*)d_in[2];   // [2,E]: src=edge[0..E), dst=edge[E..2E)
    const int*   batch = (const int*)d_in[3];
    const float* encW  = (const float*)d_in[4];
    const float* encb  = (const float*)d_in[5];
    const float* convW = (const float*)d_in[6];
    const float* convb = (const float*)d_in[7];
    const float* resW  = (const float*)d_in[8];
    const float* resb  = (const float*)d_in[9];
    const float* decW  = (const float*)d_in[10];
    const float* decb  = (const float*)d_in[11];
    float* out = (float*)d_out;

    float* ws   = (float*)d_ws;
    float* X    = ws;
    float* Y    = X   + (size_t)NPAD * NHID;
    float* hW   = Y   + (size_t)NPAD * NHID;
    float* agg  = hW  + (size_t)NPAD * NHID;
    float* dinv = agg + (size_t)NPAD * NHID;    // reused: deg -> rsqrt(deg+1)

    const int* esrc = edge;
    const int* edst = edge + NEDGES;

    dim3 blk(256);

    gcn_zero_kernel<<<(NPAD + 255) / 256, blk, 0, stream>>>(dinv, out);
    gcn_degree_kernel<<<(NEDGES + 255) / 256, blk, 0, stream>>>(edst, dinv);
    gcn_dinv_kernel<<<(NPAD + 255) / 256, blk, 0, stream>>>(dinv);

    const int gemmGrid = NPAD / 128;            // 392 blocks, 8 waves x 16 rows each
    gcn_encoder_kernel<<<gemmGrid, blk, 0, stream>>>(x, pos, encW, encb, X, Y);

    for (int l = 0; l < NLAYERS; ++l) {
        gcn_gemm_kernel<<<gemmGrid, blk, 0, stream>>>(X, convW, resW, convb, resb,
                                                      dinv, hW, agg);
        gcn_scatter_kernel<<<(NEDGES + 7) / 8, blk, 0, stream>>>(esrc, edst, hW, dinv, agg);
        gcn_update_kernel<<<(NNODES * NHID / 4 + 255) / 256, blk, 0, stream>>>(X, Y, agg);
    }

    gcn_decoder_kernel<<<(NNODES + 7) / 8, blk, 0, stream>>>(X, decW, decb, batch, out);
}